// BoundaryGaussian_68032281969159
// MI455X (gfx1250) — compile-verified
//
#include <hip/hip_runtime.h>
#include <hip/hip_bf16.h>

typedef __attribute__((ext_vector_type(2))) float     v2f;
typedef __attribute__((ext_vector_type(8))) float     v8f;
typedef __attribute__((ext_vector_type(2))) _Float16  v2h;

#define BB   8
#define HH   1024
#define WW   1024
#define RR   32
#define SEG  128
#define NTOT (BB * HH * WW)

// ---------------------------------------------------------------------------
// Kernel 1: vertical truncated EDT via two O(1) chamfer scans (exact for 1D,
// clamped at R=32 == reference's padded 65-tap min), plus global fg count.
// One thread per column; 128-row segments with 32-row halos. Forward-scan
// values kept packed 4 x u8 per dword in LDS (bank-conflict free).
// Output: g in [0,32] as u8 (8 MB instead of 32 MB fp32).
// ---------------------------------------------------------------------------
__global__ __launch_bounds__(256) void k_vert(const float* __restrict__ targets,
                                              unsigned char* __restrict__ gout,
                                              unsigned int* __restrict__ count) {
  __shared__ unsigned int gup[(192 / 4) * 256];  // 48 KB
  __shared__ unsigned int bcnt;
  const int tid = threadIdx.x;
  if (tid == 0) bcnt = 0;
  __syncthreads();

  const int x    = blockIdx.x * 256 + tid;
  const int r0   = blockIdx.y * SEG;
  const int b    = blockIdx.z;
  const int rbeg = (r0 - RR) > 0 ? (r0 - RR) : 0;                 // multiple of 4
  const int rend = (r0 + SEG + RR) < HH ? (r0 + SEG + RR) : HH;   // multiple of 4

  const float* tgt = targets + (size_t)b * HH * WW + x;

  // Forward scan: g_up[r] = fg ? 0 : min(g_up[r-1]+1, 32)   (halo init = 32)
  unsigned int pack = 0;
  int gu = RR;
  unsigned int cnt = 0;
  for (int r = rbeg; r < rend; ++r) {
    float t = tgt[(size_t)r * WW];                 // coalesced across lanes
    bool fg = t > 0.5f;
    gu = fg ? 0 : (gu < RR ? gu + 1 : RR);
    cnt += (fg && r >= r0 && r < r0 + SEG) ? 1u : 0u;
    int rl = r - rbeg;
    pack |= ((unsigned int)gu) << ((rl & 3) * 8);
    if ((rl & 3) == 3) { gup[(rl >> 2) * 256 + tid] = pack; pack = 0; }
  }

  // Backward scan: g_down and final g = min(g_up, g_down)
  int d = RR;
  unsigned int packr = 0;
  unsigned char* gorow = gout + (size_t)b * HH * WW + x;
  for (int r = rend - 1; r >= rbeg; --r) {
    int rl = r - rbeg;
    if ((rl & 3) == 3) packr = gup[(rl >> 2) * 256 + tid];
    int g_u = (int)((packr >> ((rl & 3) * 8)) & 0xFFu);
    d = (g_u == 0) ? 0 : (d < RR ? d + 1 : RR);
    int g = g_u < d ? g_u : d;
    if (r >= r0 && r < r0 + SEG)
      gorow[(size_t)r * WW] = (unsigned char)g;
  }

  atomicAdd(&bcnt, cnt);           // integer atomics: deterministic
  __syncthreads();
  if (tid == 0) atomicAdd(count, bcnt);
}

// ---------------------------------------------------------------------------
// Kernel 2: horizontal pass d2 = min_o (g^2 + o^2) in PACKED fp16 (bit-exact:
// all values are integers <= 2048), boundary weight, weighted BCE loss,
// per-block partial sum. Wave reduction via V_WMMA_F32_16X16X4_F32.
// One block per image row; thread owns 4 consecutive pixels held in two
// half2 accumulators; one dword LDS load supplies two taps; each tap feeds
// two v_pk_add_f16 + v_pk_min_f16 (2 pixels/op, o^2 are literals).
// ---------------------------------------------------------------------------
__global__ __launch_bounds__(256) void k_horiz(const float* __restrict__ probs,
                                               const unsigned char* __restrict__ gin,
                                               const unsigned int* __restrict__ count,
                                               float* __restrict__ partials) {
  __shared__ _Float16 g2h[WW + 2 * RR];   // padded row of g^2 (pad = R^2 = 1024)
  __shared__ float wsum[8];
  const int tid = threadIdx.x;
  const int r = blockIdx.x;
  const int b = blockIdx.y;
  const size_t rowoff = ((size_t)b * HH + r) * WW;

  // Stage g^2 row (+ constant pads), exactly mirroring the reference padding.
  uchar4 g4 = ((const uchar4*)(gin + rowoff))[tid];
  unsigned char gb[4] = {g4.x, g4.y, g4.z, g4.w};
#pragma unroll
  for (int k = 0; k < 4; ++k) {
    int gi = (int)gb[k];
    g2h[RR + 4 * tid + k] = (_Float16)(gi * gi);    // integer <= 1024: exact
  }
  if (tid < RR)          g2h[tid]      = (_Float16)1024.0f;  // left pad
  else if (tid < 2 * RR) g2h[WW + tid] = (_Float16)1024.0f;  // right pad
  __syncthreads();

  // Class-balance scalars (kernel 1 finished: stream-ordered).
  float S_pos = (float)(*count);
  float S_neg = (float)NTOT - S_pos;
  float w_pos = fminf((S_neg + 1e-6f) / (S_pos + 1e-6f), 1.0f);
  float w_neg_base = (S_pos + 1e-6f) / (S_neg + 1e-6f);

  // 65-tap min-plus over 4 pixels, packed two-wide. Invalid (edge) taps get a
  // large constant so the min is a no-op (reference pad semantics).
  const _Float16 BIG = (_Float16)30000.0f;
  v2h d01 = {(_Float16)1024.0f, (_Float16)1024.0f};
  v2h d23 = {(_Float16)1024.0f, (_Float16)1024.0f};
  const int base = 4 * tid;                    // even -> dword-aligned taps
#pragma unroll
  for (int t = 0; t < 68; t += 2) {
    v2h w = *(const v2h*)&g2h[base + t];       // taps t and t+1 in one dword
#pragma unroll
    for (int s = 0; s < 2; ++s) {
      const int tt = t + s;
      v2h vb = {w[s], w[s]};                   // broadcast (OPSEL-able)
      const int o0 = tt - 0 - RR, o1 = tt - 1 - RR;
      const int o2 = tt - 2 - RR, o3 = tt - 3 - RR;
      v2h c01 = {(o0 >= -RR && o0 <= RR) ? (_Float16)(float)(o0 * o0) : BIG,
                 (o1 >= -RR && o1 <= RR) ? (_Float16)(float)(o1 * o1) : BIG};
      v2h c23 = {(o2 >= -RR && o2 <= RR) ? (_Float16)(float)(o2 * o2) : BIG,
                 (o3 >= -RR && o3 <= RR) ? (_Float16)(float)(o3 * o3) : BIG};
      d01 = __builtin_elementwise_min(d01, vb + c01);   // v_pk_min / v_pk_add
      d23 = __builtin_elementwise_min(d23, vb + c23);
    }
  }
  float d2[4] = {(float)d01[0], (float)d01[1], (float)d23[0], (float)d23[1]};

  // Loss for 4 pixels; fg recovered from g==0 (no targets re-read).
  float4 pv = ((const float4*)(probs + rowoff))[tid];
  float pk[4] = {pv.x, pv.y, pv.z, pv.w};
  v2f acc = {0.0f, 0.0f};
#pragma unroll
  for (int k = 0; k < 4; ++k) {
    float pr = fminf(fmaxf(pk[k], 1e-6f), 1.0f - 1e-6f);
    float wb = expf(-d2[k] * 0.125f);                 // exp(-d2 / (2*sigma^2))
    float wneg = fminf(w_neg_base + wb, 1.0f);
    bool y = (gb[k] == 0);
    float term = y ? (w_pos * (-logf(pr))) : (wneg * (-log1pf(-pr)));
    if (k < 2) acc.x += term; else acc.y += term;
  }

  // Wave32 reduction on the matrix unit: A = lane partials (16x4 f32),
  // B = ones (4x16) -> D columns hold exact fp32 row-sums; combine M-halves.
  v2f bv = {1.0f, 1.0f};
  v8f c = {0.0f, 0.0f, 0.0f, 0.0f, 0.0f, 0.0f, 0.0f, 0.0f};
  v8f dm = __builtin_amdgcn_wmma_f32_16x16x4_f32(false, acc, false, bv,
                                                 (short)0, c, false, false);
  float s = dm[0] + dm[1] + dm[2] + dm[3] + dm[4] + dm[5] + dm[6] + dm[7];
  float tot = s + __shfl_xor(s, 16, 32);   // add other half of M rows
  int lane = tid & 31, wid = tid >> 5;
  if (lane == 0) wsum[wid] = tot;
  __syncthreads();
  if (tid == 0) {
    float t = 0.0f;
#pragma unroll
    for (int i = 0; i < 8; ++i) t += wsum[i];
    partials[(size_t)b * HH + r] = t;
  }
}

// ---------------------------------------------------------------------------
// Kernel 3: deterministic fixed-order final reduction of 8192 partials.
// ---------------------------------------------------------------------------
__global__ __launch_bounds__(256) void k_final(const float* __restrict__ partials,
                                               float* __restrict__ out) {
  __shared__ float red[256];
  const int tid = threadIdx.x;
  float s = 0.0f;
  for (int i = tid; i < BB * HH; i += 256) s += partials[i];
  red[tid] = s;
  __syncthreads();
  for (int st = 128; st > 0; st >>= 1) {
    if (tid < st) red[tid] += red[tid + st];
    __syncthreads();
  }
  if (tid == 0) out[0] = red[0] * (1.0f / (float)NTOT);
}

extern "C" void kernel_launch(void* const* d_in, const int* in_sizes, int n_in,
                              void* d_out, int out_size, void* d_ws, size_t ws_size,
                              hipStream_t stream) {
  const float* probs   = (const float*)d_in[0];
  const float* targets = (const float*)d_in[1];

  unsigned char* g        = (unsigned char*)d_ws;                       // 8 MB
  unsigned int*  count    = (unsigned int*)((char*)d_ws + (size_t)NTOT);
  float*         partials = (float*)((char*)d_ws + (size_t)NTOT + 64);  // 32 KB

  hipMemsetAsync(count, 0, sizeof(unsigned int), stream);
  k_vert <<<dim3(WW / 256, HH / SEG, BB), 256, 0, stream>>>(targets, g, count);
  k_horiz<<<dim3(HH, BB),               256, 0, stream>>>(probs, g, count, partials);
  k_final<<<1,                          256, 0, stream>>>(partials, (float*)d_out);
}